// CausalGraphPrompt_34248069218346
// MI455X (gfx1250) — compile-verified
//
#include <hip/hip_runtime.h>
#include <math.h>

// CDNA5 / gfx1250, wave32.
typedef __attribute__((ext_vector_type(2))) float v2f;
typedef __attribute__((ext_vector_type(8))) float v8f;

#define DD 128      // feature dim
#define AA 5        // anchor count

// ---------------------------------------------------------------------------
// Kernel 1: zero the agg accumulator (aliased with final_x region of d_out)
// ---------------------------------------------------------------------------
__global__ __launch_bounds__(256) void zero_agg_kernel(float4* __restrict__ p, int n4) {
    int i = blockIdx.x * 256 + threadIdx.x;
    if (i < n4) p[i] = make_float4(0.f, 0.f, 0.f, 0.f);
}

// ---------------------------------------------------------------------------
// Kernel 2: per-edge prompt + scatter-add.  One wave32 per edge; each lane
// owns 4 of the 128 feature columns.  Weights staged in LDS once per block.
// x (25.6 MB) is L2-resident, edge_prompt stores + f32 atomics dominate.
// ---------------------------------------------------------------------------
__global__ __launch_bounds__(256) void edge_kernel(
    const float* __restrict__ x,
    const float* __restrict__ anchor_edge,   // [A,D]
    const float* __restrict__ w_w,           // [2D,A]
    const float* __restrict__ w_b,           // [A]
    const int*   __restrict__ edge_index,    // [2,E]
    float*       agg,                        // [N,D] accumulate (atomic)
    float*       __restrict__ ep_out,        // [E,D]
    int N, int E)
{
    __shared__ float sWw[2 * DD * AA];   // 1280
    __shared__ float sAe[AA * DD];       // 640
    __shared__ float sWb[AA];

    for (int t = threadIdx.x; t < 2 * DD * AA; t += 256) sWw[t] = w_w[t];
    for (int t = threadIdx.x; t < AA * DD;     t += 256) sAe[t] = anchor_edge[t];
    if (threadIdx.x < AA) sWb[threadIdx.x] = w_b[threadIdx.x];
    __syncthreads();

    const int wave = threadIdx.x >> 5;
    const int lane = threadIdx.x & 31;
    const int e = blockIdx.x * 8 + wave;
    if (e >= E) return;

    const int si = edge_index[e];
    const int di = edge_index[E + e];
    const float4* x4 = (const float4*)x;
    float4 xs = x4[(long)si * 32 + lane];
    float4 xd = x4[(long)di * 32 + lane];
    const int c = lane * 4;

    // logits over A=5 anchors: dot([xs|xd], w_w[:,a])
    float p[AA];
#pragma unroll
    for (int a = 0; a < AA; ++a) {
        float s;
        s  = xs.x * sWw[(c + 0) * AA + a];
        s += xs.y * sWw[(c + 1) * AA + a];
        s += xs.z * sWw[(c + 2) * AA + a];
        s += xs.w * sWw[(c + 3) * AA + a];
        s += xd.x * sWw[(DD + c + 0) * AA + a];
        s += xd.y * sWw[(DD + c + 1) * AA + a];
        s += xd.z * sWw[(DD + c + 2) * AA + a];
        s += xd.w * sWw[(DD + c + 3) * AA + a];
        p[a] = s;
    }
#pragma unroll
    for (int off = 16; off > 0; off >>= 1) {
#pragma unroll
        for (int a = 0; a < AA; ++a) p[a] += __shfl_xor(p[a], off, 32);
    }

    // + bias, leaky_relu(0.01), softmax over A
    float mx = -INFINITY;
#pragma unroll
    for (int a = 0; a < AA; ++a) {
        float v = p[a] + sWb[a];
        v = v > 0.f ? v : 0.01f * v;
        p[a] = v;
        mx = fmaxf(mx, v);
    }
    float sum = 0.f;
#pragma unroll
    for (int a = 0; a < AA; ++a) { p[a] = expf(p[a] - mx); sum += p[a]; }
    const float inv = 1.f / sum;

    // edge_prompt = b @ anchor_edge  (this lane's 4 columns)
    float ep[4] = {0.f, 0.f, 0.f, 0.f};
#pragma unroll
    for (int a = 0; a < AA; ++a) {
        float b = p[a] * inv;
#pragma unroll
        for (int j = 0; j < 4; ++j) ep[j] += b * sAe[a * DD + c + j];
    }
    ((float4*)ep_out)[(long)e * 32 + lane] = make_float4(ep[0], ep[1], ep[2], ep[3]);

    float* as = agg + (long)si * DD + c;
    float* ad = agg + (long)di * DD + c;
#pragma unroll
    for (int j = 0; j < 4; ++j) {
        unsafeAtomicAdd(as + j, ep[j]);
        unsafeAtomicAdd(ad + j, ep[j]);
    }
}

// ---------------------------------------------------------------------------
// Kernel 3: node-side pipeline with V_WMMA_F32_16X16X4_F32.
// Block = 256 threads (8 waves) handles 32 node rows (two 16-row M-tiles).
// Wave w owns output columns [16w, 16w+16) for BOTH M-tiles, so each 4x16
// B tile loaded from L2 feeds two WMMAs (halves weight traffic vs 1 tile).
// ---------------------------------------------------------------------------
__global__ __launch_bounds__(256) void node_final_kernel(
    const float* __restrict__ x,
    const float* __restrict__ anchor_node,   // [A,D]
    const float* __restrict__ attn_w,        // [D,A]
    const float* __restrict__ attn_b,        // [A]
    const float* __restrict__ cd1_w,         // [2D,D]
    const float* __restrict__ cd1_b,
    const float* __restrict__ cd2_w,         // [D,D]
    const float* __restrict__ cd2_b,
    const float* __restrict__ int_w,         // [D,D]
    const float* __restrict__ int_b,
    float* final_agg,                        // in: agg[N,D]; out: final_x[N,D]
    int N)
{
    __shared__ float sA[32 * 256];   // [node_px | agg] per row, K=256 (32 KB)
    __shared__ float sH[32 * 128];   // relu hidden (16 KB)
    __shared__ float sS[32 * 128];   // node_px + agg (16 KB)

    const int wave = threadIdx.x >> 5;
    const int lane = threadIdx.x & 31;
    const int tile0 = blockIdx.x * 32;

    // ---- stage 1: recompute node_px for 32 rows, load agg, fill LDS ----
#pragma unroll
    for (int mm = 0; mm < 4; ++mm) {
        const int m = wave + mm * 8;            // 0..31
        int i = tile0 + m;
        const int ic = i < N ? i : N - 1;
        float4 xv = ((const float4*)x)[(long)ic * 32 + lane];
        float4 av = ((const float4*)final_agg)[(long)ic * 32 + lane];
        const int c = lane * 4;

        float p[AA];
#pragma unroll
        for (int a = 0; a < AA; ++a) {
            float s;
            s  = xv.x * attn_w[(c + 0) * AA + a];
            s += xv.y * attn_w[(c + 1) * AA + a];
            s += xv.z * attn_w[(c + 2) * AA + a];
            s += xv.w * attn_w[(c + 3) * AA + a];
            p[a] = s;
        }
#pragma unroll
        for (int off = 16; off > 0; off >>= 1) {
#pragma unroll
            for (int a = 0; a < AA; ++a) p[a] += __shfl_xor(p[a], off, 32);
        }
        float mx = -INFINITY;
#pragma unroll
        for (int a = 0; a < AA; ++a) { p[a] += attn_b[a]; mx = fmaxf(mx, p[a]); }
        float sum = 0.f;
#pragma unroll
        for (int a = 0; a < AA; ++a) { p[a] = expf(p[a] - mx); sum += p[a]; }
        const float inv = 1.f / sum;

        float xr[4] = {xv.x, xv.y, xv.z, xv.w};
        float ar[4] = {av.x, av.y, av.z, av.w};
        float px[4];
#pragma unroll
        for (int j = 0; j < 4; ++j) {
            float acc = xr[j];
#pragma unroll
            for (int a = 0; a < AA; ++a)
                acc += (p[a] * inv) * anchor_node[a * DD + c + j];
            px[j] = acc;
        }
        *((float4*)&sA[m * 256 + c])       = make_float4(px[0], px[1], px[2], px[3]);
        *((float4*)&sA[m * 256 + 128 + c]) = av;
        *((float4*)&sS[m * 128 + c]) =
            make_float4(px[0] + ar[0], px[1] + ar[1], px[2] + ar[2], px[3] + ar[3]);
    }
    __syncthreads();

    // WMMA lane mapping (16x4 f32 A / 4x16 B / 16x16 f32 C-D, wave32):
    //   A: lane = M (0..15) in both halves; VGPR v of half h holds K = 2h+v
    //   D: VGPR r: lanes 0-15 -> M=r, lanes 16-31 -> M=r+8; N = lane&15
    const int mrow = lane & 15;
    const int kk   = (lane >> 4) * 2;
    const int ncol = wave * 16 + (lane & 15);
    const int mofs = (lane >> 4) * 8;

    // ---- GEMM1: h = relu([node_px|agg] @ cd1_w + b), K = 256 ----
    v8f acc1a = {}, acc1b = {};
#pragma unroll 4
    for (int kb = 0; kb < 256; kb += 4) {
        v2f b, a0, a1;
        b.x  = cd1_w[(kb + kk) * DD + ncol];
        b.y  = cd1_w[(kb + kk + 1) * DD + ncol];
        a0.x = sA[mrow * 256 + kb + kk];
        a0.y = sA[mrow * 256 + kb + kk + 1];
        a1.x = sA[(mrow + 16) * 256 + kb + kk];
        a1.y = sA[(mrow + 16) * 256 + kb + kk + 1];
        acc1a = __builtin_amdgcn_wmma_f32_16x16x4_f32(
            false, a0, false, b, (short)0, acc1a, false, false);
        acc1b = __builtin_amdgcn_wmma_f32_16x16x4_f32(
            false, a1, false, b, (short)0, acc1b, false, false);
    }
    const float bias1 = cd1_b[ncol];
#pragma unroll
    for (int r = 0; r < 8; ++r) {
        sH[(r + mofs) * DD + ncol]        = fmaxf(acc1a[r] + bias1, 0.f);
        sH[(r + mofs + 16) * DD + ncol]   = fmaxf(acc1b[r] + bias1, 0.f);
    }
    __syncthreads();

    // ---- GEMM2: causal_strength = sigmoid(h @ cd2_w + b), K = 128 ----
    v8f acc2a = {}, acc2b = {};
#pragma unroll 4
    for (int kb = 0; kb < 128; kb += 4) {
        v2f b, a0, a1;
        b.x  = cd2_w[(kb + kk) * DD + ncol];
        b.y  = cd2_w[(kb + kk + 1) * DD + ncol];
        a0.x = sH[mrow * 128 + kb + kk];
        a0.y = sH[mrow * 128 + kb + kk + 1];
        a1.x = sH[(mrow + 16) * 128 + kb + kk];
        a1.y = sH[(mrow + 16) * 128 + kb + kk + 1];
        acc2a = __builtin_amdgcn_wmma_f32_16x16x4_f32(
            false, a0, false, b, (short)0, acc2a, false, false);
        acc2b = __builtin_amdgcn_wmma_f32_16x16x4_f32(
            false, a1, false, b, (short)0, acc2b, false, false);
    }

    // ---- GEMM3: intervened = (node_px+agg) @ int_w + b, K = 128 ----
    v8f acc3a = {}, acc3b = {};
#pragma unroll 4
    for (int kb = 0; kb < 128; kb += 4) {
        v2f b, a0, a1;
        b.x  = int_w[(kb + kk) * DD + ncol];
        b.y  = int_w[(kb + kk + 1) * DD + ncol];
        a0.x = sS[mrow * 128 + kb + kk];
        a0.y = sS[mrow * 128 + kb + kk + 1];
        a1.x = sS[(mrow + 16) * 128 + kb + kk];
        a1.y = sS[(mrow + 16) * 128 + kb + kk + 1];
        acc3a = __builtin_amdgcn_wmma_f32_16x16x4_f32(
            false, a0, false, b, (short)0, acc3a, false, false);
        acc3b = __builtin_amdgcn_wmma_f32_16x16x4_f32(
            false, a1, false, b, (short)0, acc3b, false, false);
    }

    const float b2 = cd2_b[ncol];
    const float b3 = int_b[ncol];
#pragma unroll
    for (int r = 0; r < 8; ++r) {
        const int M0 = r + mofs;
        const int i0 = tile0 + M0;
        if (i0 < N) {
            const float cs = 1.f / (1.f + expf(-(acc2a[r] + b2)));
            const float iv = acc3a[r] + b3;
            final_agg[(long)i0 * DD + ncol] = sA[M0 * 256 + ncol] + cs * iv;
        }
        const int M1 = M0 + 16;
        const int i1 = tile0 + M1;
        if (i1 < N) {
            const float cs = 1.f / (1.f + expf(-(acc2b[r] + b2)));
            const float iv = acc3b[r] + b3;
            final_agg[(long)i1 * DD + ncol] = sA[M1 * 256 + ncol] + cs * iv;
        }
    }
}

// ---------------------------------------------------------------------------
extern "C" void kernel_launch(void* const* d_in, const int* in_sizes, int n_in,
                              void* d_out, int out_size, void* d_ws, size_t ws_size,
                              hipStream_t stream) {
    const float* x           = (const float*)d_in[0];
    const float* anchor_node = (const float*)d_in[1];
    const float* attn_w      = (const float*)d_in[2];
    const float* attn_b      = (const float*)d_in[3];
    const float* anchor_edge = (const float*)d_in[4];
    const float* w_w         = (const float*)d_in[5];
    const float* w_b         = (const float*)d_in[6];
    const float* cd1_w       = (const float*)d_in[7];
    const float* cd1_b       = (const float*)d_in[8];
    const float* cd2_w       = (const float*)d_in[9];
    const float* cd2_b       = (const float*)d_in[10];
    const float* int_w       = (const float*)d_in[11];
    const float* int_b       = (const float*)d_in[12];
    const int*   edge_index  = (const int*)d_in[13];

    const int N = in_sizes[0] / DD;
    const int E = in_sizes[13] / 2;

    float* final_agg = (float*)d_out;                 // [N,D]: agg then final_x
    float* ep_out    = (float*)d_out + (long)N * DD;  // [E,D]: edge_prompt

    const int n4 = N * (DD / 4);
    zero_agg_kernel<<<(n4 + 255) / 256, 256, 0, stream>>>((float4*)final_agg, n4);

    edge_kernel<<<(E + 7) / 8, 256, 0, stream>>>(
        x, anchor_edge, w_w, w_b, edge_index, final_agg, ep_out, N, E);

    node_final_kernel<<<(N + 31) / 32, 256, 0, stream>>>(
        x, anchor_node, attn_w, attn_b,
        cd1_w, cd1_b, cd2_w, cd2_b, int_w, int_b,
        final_agg, N);
}